// MaskedPosmap2Normal_53858889892197
// MI455X (gfx1250) — compile-verified
//
#include <hip/hip_runtime.h>

#ifndef __has_builtin
#define __has_builtin(x) 0
#endif

#if __has_builtin(__builtin_amdgcn_global_load_async_to_lds_b128)
#define ASYNC_LDS 1
#else
#define ASYNC_LDS 0
#endif

#define AS_GLOBAL __attribute__((address_space(1)))
#define AS_LOCAL  __attribute__((address_space(3)))

typedef __attribute__((__vector_size__(4 * sizeof(int)))) int v4i;

#define WID 1024
#define HEI 1024
#define NBATCH 8
#define LW   264                 // staged row length in floats: columns [seg0-4, seg0+260)
#define VPR  (LW / 4)            // 66 float4 vectors per row
#define STAGE_VECS (3 * 3 * VPR) // 3 channels x 3 rows x 66 vectors = 594

__device__ __forceinline__ void wait_async_lds() {
#if ASYNC_LDS
#if __has_builtin(__builtin_amdgcn_s_wait_asynccnt)
    __builtin_amdgcn_s_wait_asynccnt(0);
#else
    asm volatile("s_wait_asynccnt 0" ::: "memory");
#endif
#endif
}

__global__ __launch_bounds__(256) void MaskedPosmap2Normal_kernel(
    const float* __restrict__ pos,          // (8,3,1024,1024) f32
    const unsigned char* __restrict__ mask, // (1024,1024) bool (1 byte)
    float* __restrict__ out)                // (8,3,1024,1024) f32
{
    // [2 buffers][3 channels][3 rows: up/center/down][LW columns]
    // Row stride = 264 floats = 1056 B (16B-aligned) -> B128-legal LDS targets.
    __shared__ float smem[2][3][3][LW];

    const int t    = threadIdx.x;
    const int i    = blockIdx.y;          // image row
    const int seg0 = blockIdx.x << 8;     // segment start column
    const int j    = seg0 + t;            // this thread's column
    const int gj0  = seg0 - 4;            // 16B-aligned global start of staged rows

    // Decompose a linear vector index into (ch, r, q).
    auto decode = [](int e, int& ch, int& r, int& q) {
        ch = e / (3 * VPR);
        int rem = e - ch * (3 * VPR);
        r = rem / VPR;
        q = rem - r * VPR;
    };
    // A staged float4 is either fully in-range or fully OOB:
    //   left halo vector covers gj=-4..-1, right halo vector covers gj=1024..1027,
    //   row-OOB (gi<0 or gi>=HEI) drops entire rows.
    auto vec_in_range = [&](int gi, int gjv) -> bool {
        return (unsigned)gi < HEI && (unsigned)gjv < (unsigned)(WID - 3);
    };

    // ---- zero-fill OOB vectors in BOTH buffers once (never async-written) ----
    for (int e = t; e < STAGE_VECS; e += 256) {
        int ch, r, q; decode(e, ch, r, q);
        int gi  = i + r - 1;
        int gjv = gj0 + q * 4;
        if (!vec_in_range(gi, gjv)) {
            float* l0 = &smem[0][ch][r][q * 4];
            float* l1 = &smem[1][ch][r][q * 4];
            l0[0] = 0.0f; l0[1] = 0.0f; l0[2] = 0.0f; l0[3] = 0.0f;
            l1[0] = 0.0f; l1[1] = 0.0f; l1[2] = 0.0f; l1[3] = 0.0f;
        }
    }

    // ---- mask validity weights: depend only on (i,j); computed once, reused 8x ----
    auto mval = [&](int mi, int mj) -> float {
        if ((unsigned)mi >= HEI || (unsigned)mj >= WID) return 0.0f;
        return mask[mi * WID + mj] ? 1.0f : 0.0f;
    };
    const float mc = mval(i, j);
    const float mu = mval(i - 1, j);
    const float md = mval(i + 1, j);
    const float ml = mval(i, j - 1);
    const float mr = mval(i, j + 1);
    const float v0 = (mu + mc + mr == 3.0f) ? 1.0f : 0.0f; // up+c+right
    const float v1 = (mu + ml + mc == 3.0f) ? 1.0f : 0.0f; // up+left+c
    const float v2 = (ml + mc + md == 3.0f) ? 1.0f : 0.0f; // left+c+down
    const float v3 = (mc + mr + md == 3.0f) ? 1.0f : 0.0f; // c+right+down

    // ---- stage one batch's 3ch x 3row halo tile into LDS buffer `buf` (B128 async) ----
    auto stage = [&](int b, int buf) {
        for (int e = t; e < STAGE_VECS; e += 256) {
            int ch, r, q; decode(e, ch, r, q);
            int gi  = i + r - 1;
            int gjv = gj0 + q * 4;
            if (vec_in_range(gi, gjv)) {
                const float* gp = pos + (((size_t)(b * 3 + ch) * HEI + gi) * WID + gjv);
                float* lp = &smem[buf][ch][r][q * 4];
#if ASYNC_LDS
                __builtin_amdgcn_global_load_async_to_lds_b128(
                    (AS_GLOBAL v4i*)gp, (AS_LOCAL v4i*)lp, 0, 0);
#else
                *(float4*)lp = *(const float4*)gp;
#endif
            }
        }
    };

    // ---- prologue: stage batch 0, then double-buffer through the batch loop ----
    stage(0, 0);
    wait_async_lds();
    __syncthreads();

    const int lc = t + 4;  // this thread's center column within the staged row
    for (int b = 0; b < NBATCH; ++b) {
        const int buf = b & 1;
        if (b + 1 < NBATCH) stage(b + 1, buf ^ 1);  // async prefetch of next batch

        // 5-point stencil gathered from LDS
        float cc[3], rr[3], ll[3], uu[3], dd[3];
#pragma unroll
        for (int ch = 0; ch < 3; ++ch) {
            const float* row1 = &smem[buf][ch][1][0];
            cc[ch] = row1[lc];
            rr[ch] = row1[lc + 1];
            ll[ch] = row1[lc - 1];
            uu[ch] = smem[buf][ch][0][lc];
            dd[ch] = smem[buf][ch][2][lc];
        }
        float d0[3], d1[3], d2[3], d3[3];
#pragma unroll
        for (int ch = 0; ch < 3; ++ch) {
            d0[ch] = rr[ch] - cc[ch];   // right - c
            d1[ch] = uu[ch] - cc[ch];   // up    - c
            d2[ch] = ll[ch] - cc[ch];   // left  - c
            d3[ch] = dd[ch] - cc[ch];   // down  - c
        }
        auto CX = [](const float* a, const float* bb, float* o) {
            o[0] = a[1] * bb[2] - a[2] * bb[1];
            o[1] = a[2] * bb[0] - a[0] * bb[2];
            o[2] = a[0] * bb[1] - a[1] * bb[0];
        };
        float c01[3], c12[3], c23[3], c30[3];
        CX(d0, d1, c01); CX(d1, d2, c12); CX(d2, d3, c23); CX(d3, d0, c30);

        float n0 = v0 * c01[0] + v1 * c12[0] + v2 * c23[0] + v3 * c30[0];
        float n1 = v0 * c01[1] + v1 * c12[1] + v2 * c23[1] + v3 * c30[1];
        float n2 = v0 * c01[2] + v1 * c12[2] + v2 * c23[2] + v3 * c30[2];

        float len = sqrtf(n0 * n0 + n1 * n1 + n2 * n2);
        float inv = 1.0f / fmaxf(len, 1e-12f);

        out[((size_t)(b * 3 + 0) * HEI + i) * WID + j] = n0 * inv;
        out[((size_t)(b * 3 + 1) * HEI + i) * WID + j] = n1 * inv;
        out[((size_t)(b * 3 + 2) * HEI + i) * WID + j] = n2 * inv;

        wait_async_lds();   // this wave's async fills of the other buffer have landed
        __syncthreads();    // all waves: fills visible, current buffer fully consumed
    }
}

extern "C" void kernel_launch(void* const* d_in, const int* in_sizes, int n_in,
                              void* d_out, int out_size, void* d_ws, size_t ws_size,
                              hipStream_t stream) {
    const float* pos          = (const float*)d_in[0];
    const unsigned char* mask = (const unsigned char*)d_in[1];
    float* out                = (float*)d_out;
    (void)in_sizes; (void)n_in; (void)out_size; (void)d_ws; (void)ws_size;

    dim3 grid(WID / 256, HEI, 1);  // (4, 1024): one row-segment per block, batches looped inside
    MaskedPosmap2Normal_kernel<<<grid, dim3(256, 1, 1), 0, stream>>>(pos, mask, out);
}